// LayerGIN_12893491823105
// MI455X (gfx1250) — compile-verified
//
#include <hip/hip_runtime.h>
#include <hip/hip_bf16.h>

// GIN layer for MI455X (gfx1250, wave32).
// Pipeline: agg = SpMM(edges, v) + eps*v ; y1 = relu(BN(agg@W1+b1)) ; out = relu(BN(y1@W2+b2))
// fp32 throughout; GEMMs use V_WMMA_F32_16X16X4_F32 (exact-precision fp32 matrix core path).

typedef float v2f __attribute__((ext_vector_type(2)));
typedef float v8f __attribute__((ext_vector_type(8)));

#define GIN_D_IN  128
#define GIN_D_HID 256
#define GIN_D_OUT 128
#define GIN_BN_EPS 1e-5f

// ---------------- init: agg = eps * v, zero BN stat accumulators ----------------
__global__ void gin_init_kernel(const float* __restrict__ v,
                                const float* __restrict__ eps,
                                float* __restrict__ agg,
                                float* __restrict__ stats, int nstats,
                                long total) {
    long stride = (long)gridDim.x * blockDim.x;
    long i0 = (long)blockIdx.x * blockDim.x + threadIdx.x;
    float e = eps[0];
    for (long i = i0; i < total; i += stride) agg[i] = e * v[i];
    if (i0 < nstats) stats[i0] = 0.0f;
}

// ---------------- SpMM: one wave per edge, float4 per lane, HW f32 atomics ------
__global__ void gin_spmm_kernel(const float* __restrict__ v,
                                const int* __restrict__ rows,
                                const int* __restrict__ cols,
                                const float* __restrict__ vals,
                                float* __restrict__ agg, int E) {
    int e = blockIdx.x * (blockDim.x >> 5) + (threadIdx.x >> 5);
    if (e >= E) return;
    int lane = threadIdx.x & 31;
    int r = rows[e];
    int c = cols[e];
    float w = vals[e];
    // 32 lanes x float4 = 128 floats = one node row
    float4 m = ((const float4*)(v + (long)c * GIN_D_IN))[lane];
    float* dst = agg + (long)r * GIN_D_IN + lane * 4;
    __hip_atomic_fetch_add(dst + 0, w * m.x, __ATOMIC_RELAXED, __HIP_MEMORY_SCOPE_AGENT);
    __hip_atomic_fetch_add(dst + 1, w * m.y, __ATOMIC_RELAXED, __HIP_MEMORY_SCOPE_AGENT);
    __hip_atomic_fetch_add(dst + 2, w * m.z, __ATOMIC_RELAXED, __HIP_MEMORY_SCOPE_AGENT);
    __hip_atomic_fetch_add(dst + 3, w * m.w, __ATOMIC_RELAXED, __HIP_MEMORY_SCOPE_AGENT);
}

// ---------------- WMMA fp32 GEMM + bias + column-stat accumulation -------------
// Block = 128 threads (4 waves). Block computes 16 rows x NOUT cols; each wave a
// 16 x (NOUT/4) strip via V_WMMA_F32_16X16X4_F32 chained over K.
// Requires Nrows % 16 == 0 (100000 = 6250*16).
template <int K, int NOUT>
__global__ void gin_gemm_wmma_kernel(const float* __restrict__ X,
                                     const float* __restrict__ W,
                                     const float* __restrict__ bias,
                                     float* __restrict__ Y,
                                     float* __restrict__ csum,
                                     float* __restrict__ csq) {
    constexpr int WAVES = 4;
    constexpr int TN = NOUT / WAVES;  // columns per wave
    constexpr int NT = TN / 16;       // 16-wide subtiles per wave
    const int wave = threadIdx.x >> 5;
    const int lane = threadIdx.x & 31;
    const int l15 = lane & 15;
    const int hi  = lane >> 4;        // 0: K=k0..k0+1 ; 1: K=k0+2..k0+3
    const int m0  = blockIdx.x * 16;
    const int row = m0 + l15;         // A-matrix row held by this lane
    const int n0  = wave * TN;

    v8f c[NT] = {};  // 16x16 f32 accumulators (8 VGPRs each)

    const float* __restrict__ xrow = X + (long)row * K + hi * 2;
    for (int k0 = 0; k0 < K; k0 += 4) {
        // A 16x4 fragment: lane<16 -> {K=k0,k0+1}, lane>=16 -> {K=k0+2,k0+3}
        v2f a = *(const v2f*)(xrow + k0);
        const float* wk = W + (long)(k0 + hi * 2) * NOUT;
#pragma unroll
        for (int t = 0; t < NT; ++t) {
            int n = n0 + t * 16 + l15;
            v2f b;
            b.x = wk[n];          // B[k0+hi*2    ][n]
            b.y = wk[NOUT + n];   // B[k0+hi*2 + 1][n]
            c[t] = __builtin_amdgcn_wmma_f32_16x16x4_f32(
                false, a, false, b, (short)0, c[t], false, false);
        }
    }

    const float invalid = 0.0f; (void)invalid;
#pragma unroll
    for (int t = 0; t < NT; ++t) {
        int n = n0 + t * 16 + l15;
        float bv = bias[n];
        float s = 0.0f, q = 0.0f;
#pragma unroll
        for (int r8 = 0; r8 < 8; ++r8) {
            // C layout: VGPR r8, lane<16 -> row m0+r8, lane>=16 -> row m0+r8+8
            float val = c[t][r8] + bv;
            Y[(long)(m0 + r8 + hi * 8) * NOUT + n] = val;
            s += val;
            q += val * val;
        }
        // combine the two half-wave row groups (same column n)
        s += __shfl_xor(s, 16, 32);
        q += __shfl_xor(q, 16, 32);
        if (hi == 0) {
            __hip_atomic_fetch_add(csum + n, s, __ATOMIC_RELAXED, __HIP_MEMORY_SCOPE_AGENT);
            __hip_atomic_fetch_add(csq + n, q, __ATOMIC_RELAXED, __HIP_MEMORY_SCOPE_AGENT);
        }
    }
}

// ---------------- fold BN stats + gamma/beta into per-column scale/shift -------
__global__ void gin_bn_stats_kernel(const float* __restrict__ csum,
                                    const float* __restrict__ csq,
                                    const float* __restrict__ gamma,
                                    const float* __restrict__ beta,
                                    float* __restrict__ scale,
                                    float* __restrict__ shift,
                                    int nout, float invN) {
    int i = blockIdx.x * blockDim.x + threadIdx.x;
    if (i >= nout) return;
    float mu  = csum[i] * invN;
    float var = csq[i] * invN - mu * mu;
    float rstd = rsqrtf(var + GIN_BN_EPS);
    float sc = gamma[i] * rstd;
    scale[i] = sc;
    shift[i] = beta[i] - mu * sc;
}

// ---------------- elementwise BN + ReLU (in place) -----------------------------
template <int NOUT>
__global__ void gin_bn_relu_kernel(float* __restrict__ y,
                                   const float* __restrict__ scale,
                                   const float* __restrict__ shift,
                                   long total) {
    long stride = (long)gridDim.x * blockDim.x;
    for (long i = (long)blockIdx.x * blockDim.x + threadIdx.x; i < total; i += stride) {
        int col = (int)(i & (NOUT - 1));
        y[i] = fmaxf(0.0f, y[i] * scale[col] + shift[col]);
    }
}

extern "C" void kernel_launch(void* const* d_in, const int* in_sizes, int n_in,
                              void* d_out, int out_size, void* d_ws, size_t ws_size,
                              hipStream_t stream) {
    // setup_inputs() order
    const float* v        = (const float*)d_in[0];
    const int*   erows    = (const int*)d_in[1];
    const int*   ecols    = (const int*)d_in[2];
    const float* evals    = (const float*)d_in[3];
    const float* eps      = (const float*)d_in[4];
    const float* W1       = (const float*)d_in[5];
    const float* b1       = (const float*)d_in[6];
    const float* g1       = (const float*)d_in[7];
    const float* be1      = (const float*)d_in[8];
    const float* W2       = (const float*)d_in[9];
    const float* b2       = (const float*)d_in[10];
    const float* g2       = (const float*)d_in[11];
    const float* be2      = (const float*)d_in[12];
    float* out = (float*)d_out;

    const int N = in_sizes[0] / GIN_D_IN;   // 100000
    const int E = in_sizes[1];              // 1600000

    // Workspace layout (floats)
    float* ws = (float*)d_ws;
    float* agg = ws;                                   // N * 128
    float* y1  = agg + (long)N * GIN_D_IN;             // N * 256
    float* st  = y1 + (long)N * GIN_D_HID;             // stats block
    float* sum1   = st;           // 256
    float* sq1    = st + 256;     // 256
    float* scale1 = st + 512;     // 256
    float* shift1 = st + 768;     // 256
    float* sum2   = st + 1024;    // 128
    float* sq2    = st + 1152;    // 128
    float* scale2 = st + 1280;    // 128
    float* shift2 = st + 1408;    // 128
    const int NSTATS = 768;       // zeroes st[0..767] = sum1,sq1,scale1; sum2,sq2 not zeroed (768 <= 1 block)

    // 1) agg = eps * v ; zero stat accumulators
    long aggTotal = (long)N * GIN_D_IN;
    gin_init_kernel<<<4096, 1024, 0, stream>>>(v, eps, agg, sum1, NSTATS, aggTotal);

    // 2) SpMM scatter-add: one wave per edge (8 edges per 256-thread block)
    int spmmBlocks = (E + 7) / 8;
    gin_spmm_kernel<<<spmmBlocks, 256, 0, stream>>>(v, erows, ecols, evals, agg, E);

    // 3) y1_pre = agg @ W1 + b1 (WMMA fp32), accumulate column sum/sumsq
    gin_gemm_wmma_kernel<GIN_D_IN, GIN_D_HID>
        <<<N / 16, 128, 0, stream>>>(agg, W1, b1, y1, sum1, sq1);

    // 4) BN stats 1
    gin_bn_stats_kernel<<<1, GIN_D_HID, 0, stream>>>(sum1, sq1, g1, be1, scale1, shift1,
                                                     GIN_D_HID, 1.0f / (float)N);

    // 5) y1 = relu(BN(y1_pre)) in place
    gin_bn_relu_kernel<GIN_D_HID><<<8192, 256, 0, stream>>>(y1, scale1, shift1,
                                                            (long)N * GIN_D_HID);

    // 6) out_pre = y1 @ W2 + b2 (WMMA fp32) into d_out, accumulate stats
    gin_gemm_wmma_kernel<GIN_D_HID, GIN_D_OUT>
        <<<N / 16, 128, 0, stream>>>(y1, W2, b2, out, sum2, sq2);

    // 7) BN stats 2
    gin_bn_stats_kernel<<<1, GIN_D_OUT, 0, stream>>>(sum2, sq2, g2, be2, scale2, shift2,
                                                     GIN_D_OUT, 1.0f / (float)N);

    // 8) out = relu(BN(out_pre)) in place
    gin_bn_relu_kernel<GIN_D_OUT><<<8192, 256, 0, stream>>>(out, scale2, shift2,
                                                            (long)N * GIN_D_OUT);
}